// SpatialCorrelationSampler_23845658427551
// MI455X (gfx1250) — compile-verified
//
#include <hip/hip_runtime.h>

// 3D spatial correlation sampler (patch=7, dil=1, k=1, s=1, p=0) for MI455X.
// out[b,pd,ph,pw,d,h,w] = sum_c in1[b,c,d,h,w] * in2[b,c,d+pd-3,h+ph-3,w+pw-3]
// (zero-padded in2). Store-bound (90MB out); C-reduction done with
// V_WMMA_F32_16X16X4_F32 as a banded GEMM over 16-wide w tiles; band is
// re-coalesced through LDS before the global stores.

typedef __attribute__((ext_vector_type(2))) float v2f;
typedef __attribute__((ext_vector_type(8))) float v8f;

#define PATCH 7
#define RAD   3
#define CC    32
#define DD    32
#define HH    32
#define WW    32
#define DHW   (DD*HH*WW)        // 32768
#define CDHW  (CC*DHW)

__global__ __launch_bounds__(32)
void corr3d_wmma_kernel(const float* __restrict__ in1,
                        const float* __restrict__ in2,
                        float* __restrict__ out)
{
    __shared__ float sbuf[PATCH * 16];   // band staging: [pw][i]

    const int lane = threadIdx.x;        // wave32
    const int jb   = lane & 15;          // N column within chunk / M row for A
    const int half = lane >> 4;          // 0: lanes 0-15, 1: lanes 16-31

    int bid = blockIdx.x;
    const int pd = bid % PATCH; bid /= PATCH;
    const int wt = bid & 1;     bid >>= 1;
    const int h  = bid % HH;    bid /= HH;
    const int d  = bid % DD;    bid /= DD;
    const int b  = bid;
    const int w0 = wt * 16;

    // ---- A tile: in1[b, 0..31, d, h, w0+jb]; WMMA f32 A layout:
    // K-step k uses VGPR v holding channel c = 4k + 2*half + v.
    float areg[16];
    {
        const float* p1 = in1 + (size_t)b * CDHW + (size_t)d * (HH*WW)
                              + (size_t)h * WW + (w0 + jb);
        #pragma unroll
        for (int k = 0; k < 8; ++k) {
            areg[2*k+0] = p1[(size_t)(4*k + 2*half + 0) * DHW];
            areg[2*k+1] = p1[(size_t)(4*k + 2*half + 1) * DHW];
        }
    }

    const int  dz  = d + pd - RAD;
    const bool dok = (unsigned)dz < (unsigned)DD;
    const int  dzc = min(max(dz, 0), DD - 1);

    for (int ph = 0; ph < PATCH; ++ph) {
        const int  hy    = h + ph - RAD;
        const bool rowok = dok && ((unsigned)hy < (unsigned)HH);
        const int  hyc   = min(max(hy, 0), HH - 1);

        // B columns j: chunk0 j=jb, chunk1 j=jb+16; in2 w index wx = w0-3+j.
        const int wx0 = w0 - RAD + jb;
        const int wx1 = wx0 + 16;
        // Branchless zero padding: clamp address, scale by 0/1 mask
        // (keeps EXEC all-ones for WMMA, keeps loads coalesced).
        const float m0 = (rowok && (unsigned)wx0 < (unsigned)WW) ? 1.0f : 0.0f;
        const float m1 = (rowok && (unsigned)wx1 < (unsigned)WW && jb < (PATCH - 1))
                             ? 1.0f : 0.0f;
        const int wx0c = min(max(wx0, 0), WW - 1);
        const int wx1c = min(max(wx1, 0), WW - 1);

        const float* p2  = in2 + (size_t)b * CDHW + (size_t)dzc * (HH*WW)
                               + (size_t)hyc * WW;
        const float* p20 = p2 + wx0c;
        const float* p21 = p2 + wx1c;

        v8f acc0 = {};
        v8f acc1 = {};

        #pragma unroll
        for (int k = 0; k < 8; ++k) {
            v2f a;
            a[0] = areg[2*k + 0];
            a[1] = areg[2*k + 1];
            const size_t c0 = (size_t)(4*k + 2*half + 0) * DHW;
            const size_t c1 = (size_t)(4*k + 2*half + 1) * DHW;
            v2f b0, b1;
            b0[0] = m0 * p20[c0];  b0[1] = m0 * p20[c1];
            b1[0] = m1 * p21[c0];  b1[1] = m1 * p21[c1];
            // D = A(16x4) x B(4x16) + C, fp32
            acc0 = __builtin_amdgcn_wmma_f32_16x16x4_f32(
                       false, a, false, b0, (short)0, acc0, false, false);
            acc1 = __builtin_amdgcn_wmma_f32_16x16x4_f32(
                       false, a, false, b1, (short)0, acc1, false, false);
        }

        // Band extraction: D[i,j] -> out(w=w0+i, pw=j-i), valid pw in [0,7).
        // D layout: VGPR r holds M = r + 8*half, N = jb (+16 for chunk1).
        #pragma unroll
        for (int r = 0; r < 8; ++r) {
            const int i = r + 8 * half;
            int pw = jb - i;
            if (pw >= 0 && pw < PATCH) sbuf[pw * 16 + i] = acc0[r];
            pw = jb + 16 - i;
            if (pw >= 0 && pw < PATCH) sbuf[pw * 16 + i] = acc1[r];
        }
        __syncthreads();

        // Coalesced stores: per pw plane, 16 contiguous w (64B per half-wave).
        float* obase = out
            + ((((size_t)b * PATCH + pd) * PATCH + ph) * PATCH) * DHW
            + (size_t)d * (HH*WW) + (size_t)h * WW + w0;
        #pragma unroll
        for (int e0 = 0; e0 < PATCH * 16; e0 += 32) {
            const int e = e0 + lane;
            if (e < PATCH * 16) {
                const int pw = e >> 4;
                const int i  = e & 15;
                obase[(size_t)pw * DHW + i] = sbuf[e];
            }
        }
        __syncthreads();
    }
}

extern "C" void kernel_launch(void* const* d_in, const int* in_sizes, int n_in,
                              void* d_out, int out_size, void* d_ws, size_t ws_size,
                              hipStream_t stream) {
    (void)in_sizes; (void)n_in; (void)d_ws; (void)ws_size; (void)out_size;
    const float* in1 = (const float*)d_in[0];
    const float* in2 = (const float*)d_in[1];
    float* out = (float*)d_out;

    // grid: B(2) * D(32) * H(32) * Wtiles(2) * pd(7) = 28672 single-wave WGs
    const int nblocks = 2 * 32 * 32 * 2 * PATCH;
    corr3d_wmma_kernel<<<dim3(nblocks), dim3(32), 0, stream>>>(in1, in2, out);
}